// CombinationalCircuit_31911607009919
// MI455X (gfx1250) — compile-verified
//
#include <hip/hip_runtime.h>
#include <hip/hip_bf16.h>
#include <stdint.h>

// ---------------------------------------------------------------------------
// 3-SAT probabilistic circuit evaluation on MI455X (gfx1250).
//
//   out[b] = prod_c ( 1 - prod_k (1 - lit(b,c,k)) ),  lit = sign ? p : 1-p,
//   p = sigmoid(emb[b][var])
//
// Memory-bound ragged gather. Strategy:
//   k1: xT[v][b] = sigmoid(emb[b][v])  (LDS-tiled transpose, 25.6MB, L2-resident)
//   k2: per-block clause range; clause metadata DMA'd into LDS by the CDNA5
//       Tensor Data Mover with ping-pong double buffering (tensor_load_to_lds
//       + in-order TENSORcnt: wait<=2 overlaps next-chunk DMA with compute);
//       gathers become contiguous 256B row loads of xT; per-thread product.
//   k3: product-reduce per-block partials -> out[64].
// WMMA is not applicable: there is no contraction dimension in this op.
// ---------------------------------------------------------------------------

typedef unsigned int u32;
typedef unsigned long long u64;

typedef __attribute__((ext_vector_type(4))) u32 v4u;
typedef __attribute__((ext_vector_type(8))) int  v8i;
typedef __attribute__((ext_vector_type(4))) int  v4i;

#define BATCH 64
#define NSUB  4        // clause sub-groups per block (blockDim = 64 x 4 = 8 waves)
#define CHUNK 256      // clauses staged into LDS per TDM transfer

#if defined(__HIP_DEVICE_COMPILE__) && defined(__gfx1250__) && \
    __has_builtin(__builtin_amdgcn_tensor_load_to_lds)
#define USE_TDM 1
#else
#define USE_TDM 0
#endif

__device__ __forceinline__ u32 lds_byte_offset(const void* p) {
#if defined(__HIP_DEVICE_COMPILE__)
  // ptrtoint of an LDS (addrspace(3)) pointer yields the byte offset used by
  // DS/TDM instructions.
  return (u32)(uintptr_t)(__attribute__((address_space(3))) const char*)p;
#else
  (void)p;
  return 0u;
#endif
}

#if USE_TDM
// Issue a 1-D TDM load of `ndw` dwords from global `gptr` into LDS at
// byte offset `lds_off`. Descriptor bitfields per CDNA5 ISA §8.3/§8.4.
// Call from exactly one wave; fence with s_wait_tensorcnt.
__device__ __forceinline__ void tdm_load_dwords_to_lds(u32 lds_off,
                                                       const void* gptr,
                                                       u32 ndw) {
  u64 ga = (u64)(uintptr_t)gptr;
  v4u g0;
  g0.x = 0x1u;                                           // count=1, user D#
  g0.y = lds_off;                                        // lds_addr (bytes)
  g0.z = (u32)ga;                                        // global_addr[31:0]
  g0.w = ((u32)(ga >> 32) & 0x01FFFFFFu) | (2u << 30);   // ga[56:32] | type=2

  v8i g1;
  g1[0] = (int)(2u << 16);                               // data_size = 4 bytes
  g1[1] = (int)((ndw & 0xFFFFu) << 16);                  // tensor_dim0[15:0]
  g1[2] = (int)(((ndw >> 16) & 0xFFFFu) | (1u << 16));   // dim0[31:16] | tensor_dim1=1
  g1[3] = (int)((ndw & 0xFFFFu) << 16);                  // tile_dim0 = ndw (<=65535)
  g1[4] = 0;                                             // tile_dim1 = tile_dim2 = 0
  g1[5] = (int)ndw;                                      // tensor_dim0_stride[31:0]
  g1[6] = 0;                                             // stride0 hi / stride1 lo
  g1[7] = 0;                                             // stride1 hi

  v4i gz4 = {0, 0, 0, 0};                                // groups 2/3 unused (<=2D)
  v8i gz8 = {0, 0, 0, 0, 0, 0, 0, 0};                    // extra group slot unused
  __builtin_amdgcn_tensor_load_to_lds(g0, g1, gz4, gz4, gz8, /*cpol=*/0);
}
#endif

// ---------------------------------------------------------------------------
// Kernel 1: xT[v*64 + b] = sigmoid(emb[b*NV + v]); LDS-tiled 64x64 transpose so
// both the global read and the global write are fully coalesced.
// ---------------------------------------------------------------------------
__global__ void __launch_bounds__(256)
sigmoid_transpose_kernel(const float* __restrict__ emb,
                         float* __restrict__ xT,
                         int NV) {
  __shared__ float tile[64][65];                 // +1 pad: conflict-free transpose
  const int tx = threadIdx.x;                    // 0..63
  const int ty = threadIdx.y;                    // 0..3
  const int v0 = blockIdx.x * 64;

  #pragma unroll
  for (int i = 0; i < 16; ++i) {
    const int b = ty * 16 + i;
    const int v = v0 + tx;
    tile[b][tx] = (v < NV) ? emb[(size_t)b * (size_t)NV + (size_t)v] : 0.0f;
  }
  __syncthreads();
  #pragma unroll
  for (int i = 0; i < 16; ++i) {
    const int vl = ty * 16 + i;
    const int v  = v0 + vl;
    if (v < NV) {
      const float e = tile[tx][vl];              // b = tx
      const float s = 1.0f / (1.0f + __expf(-e));
      xT[(size_t)v * BATCH + tx] = s;
    }
  }
}

// ---------------------------------------------------------------------------
// Kernel 2: each block evaluates a contiguous clause range. Clause metadata is
// double-buffer DMA'd into LDS by the Tensor Data Mover (next chunk in flight
// while the current one is consumed); each 64-thread group (b = tx) evaluates
// one clause via three contiguous 256B row loads of xT.
// ---------------------------------------------------------------------------
__global__ void __launch_bounds__(256)
clause_eval_kernel(const float* __restrict__ xT,
                   const int* __restrict__ cvars,
                   const int* __restrict__ csigns,
                   float* __restrict__ partials,
                   int C) {
  __shared__ int   s_vars [2][CHUNK * 3];        // ping-pong buffers
  __shared__ int   s_signs[2][CHUNK * 3];
  __shared__ float s_acc[NSUB][BATCH];

  const int tx = threadIdx.x;                    // batch lane, 0..63
  const int ty = threadIdx.y;                    // clause sub-group, 0..3
  const int nb = (int)gridDim.x;

  const int per  = (C + nb - 1) / nb;
  const int cbeg = (int)blockIdx.x * per;
  const int cend = (cbeg + per < C) ? (cbeg + per) : C;
  const int span = cend - cbeg;
  const int nchunks = (span > 0) ? (span + CHUNK - 1) / CHUNK : 0;

#if USE_TDM
  const bool wave0 = (ty == 0 && tx < 32);       // wave-uniform predicate
  if (wave0 && nchunks > 0) {                    // prologue: DMA chunk 0
    const int n0 = (span < CHUNK) ? span : CHUNK;
    tdm_load_dwords_to_lds(lds_byte_offset(s_vars[0]),
                           cvars + (size_t)cbeg * 3, (u32)(n0 * 3));
    tdm_load_dwords_to_lds(lds_byte_offset(s_signs[0]),
                           csigns + (size_t)cbeg * 3, (u32)(n0 * 3));
  }
#endif

  float prod = 1.0f;
  int buf = 0;

  for (int ci = 0; ci < nchunks; ++ci) {
    const int base = cbeg + ci * CHUNK;
    const int n    = ((cend - base) < CHUNK) ? (cend - base) : CHUNK;

#if USE_TDM
    if (wave0) {
      if (ci + 1 < nchunks) {                    // kick off next chunk's DMA
        const int nbase = base + CHUNK;
        const int nn    = ((cend - nbase) < CHUNK) ? (cend - nbase) : CHUNK;
        tdm_load_dwords_to_lds(lds_byte_offset(s_vars[buf ^ 1]),
                               cvars + (size_t)nbase * 3, (u32)(nn * 3));
        tdm_load_dwords_to_lds(lds_byte_offset(s_signs[buf ^ 1]),
                               csigns + (size_t)nbase * 3, (u32)(nn * 3));
        // Tensor ops complete in order per wave: <=2 outstanding means the
        // current chunk's two transfers have landed; next two still in flight.
        __builtin_amdgcn_s_wait_tensorcnt(2);
      } else {
        __builtin_amdgcn_s_wait_tensorcnt(0);
      }
    }
#else
    for (int i = ty * 64 + tx; i < n * 3; i += 256) {
      s_vars[buf][i]  = cvars [(size_t)base * 3 + i];
      s_signs[buf][i] = csigns[(size_t)base * 3 + i];
    }
#endif
    __syncthreads();                             // publish current chunk

    for (int cl = ty; cl < n; cl += NSUB) {
      const int v0 = s_vars[buf][cl * 3 + 0];
      const int v1 = s_vars[buf][cl * 3 + 1];
      const int v2 = s_vars[buf][cl * 3 + 2];
      const int g0 = s_signs[buf][cl * 3 + 0];
      const int g1 = s_signs[buf][cl * 3 + 1];
      const int g2 = s_signs[buf][cl * 3 + 2];

      const float p0 = xT[(size_t)v0 * BATCH + tx];
      const float p1 = xT[(size_t)v1 * BATCH + tx];
      const float p2 = xT[(size_t)v2 * BATCH + tx];

      const float l0 = (g0 > 0) ? p0 : 1.0f - p0;
      const float l1 = (g1 > 0) ? p1 : 1.0f - p1;
      const float l2 = (g2 > 0) ? p2 : 1.0f - p2;

      const float csat = 1.0f - (1.0f - l0) * (1.0f - l1) * (1.0f - l2);
      prod *= csat;
    }
    __syncthreads();                             // done consuming this buffer
    buf ^= 1;
  }

  s_acc[ty][tx] = prod;
  __syncthreads();
  if (ty == 0) {
    const float r = s_acc[0][tx] * s_acc[1][tx] * s_acc[2][tx] * s_acc[3][tx];
    partials[(size_t)blockIdx.x * BATCH + tx] = r;
  }
}

// ---------------------------------------------------------------------------
// Kernel 3: out[b] = prod over blocks of partials[block][b].
// ---------------------------------------------------------------------------
__global__ void __launch_bounds__(64)
reduce_product_kernel(const float* __restrict__ partials,
                      float* __restrict__ out,
                      int nb) {
  const int tx = threadIdx.x;                    // 0..63
  float p = 1.0f;
  for (int i = 0; i < nb; ++i) {
    p *= partials[(size_t)i * BATCH + tx];
  }
  out[tx] = p;
}

// ---------------------------------------------------------------------------
// Host-side launcher.
// inputs: [0] input_idx (identity, unused), [1] emb_weight f32 [64, NV],
//         [2] clause_vars i32 [C,3], [3] clause_signs i32 [C,3]
// output: f32 [64]
// ws layout: xT (NV*64 f32) | partials (NB*64 f32)
// ---------------------------------------------------------------------------
extern "C" void kernel_launch(void* const* d_in, const int* in_sizes, int n_in,
                              void* d_out, int out_size, void* d_ws, size_t ws_size,
                              hipStream_t stream) {
  (void)n_in; (void)out_size;

  const float* emb    = (const float*)d_in[1];
  const int*   cvars  = (const int*)d_in[2];
  const int*   csigns = (const int*)d_in[3];

  const int NV = in_sizes[1] / BATCH;            // 100000
  const int C  = in_sizes[2] / 3;                // 400000

  float* xT = (float*)d_ws;
  const size_t xt_elems = (size_t)NV * BATCH;
  float* partials = xT + xt_elems;

  // Cap block count so the partial-product buffer fits in the workspace.
  int NB = 1024;
  const size_t avail = ws_size / sizeof(float);
  if (avail > xt_elems) {
    const size_t maxnb = (avail - xt_elems) / BATCH;
    if ((size_t)NB > maxnb) NB = (int)maxnb;
  }
  if (NB < 1) NB = 1;
  if (NB > C) NB = C;

  dim3 blk(64, NSUB);

  sigmoid_transpose_kernel<<<(NV + 63) / 64, blk, 0, stream>>>(emb, xT, NV);
  clause_eval_kernel<<<NB, blk, 0, stream>>>(xT, cvars, csigns, partials, C);
  reduce_product_kernel<<<1, 64, 0, stream>>>(partials, (float*)d_out, NB);
}